// EfficientDedispersionLayer_68745246540395
// MI455X (gfx1250) — compile-verified
//
#include <hip/hip_runtime.h>
#include <hip/hip_bf16.h>

// ---------------------------------------------------------------------------
// EfficientDedispersionLayer for MI455X (gfx1250, wave32)
//
//   out[b,d,p,t,f] = x[b,p,(t + shift[d, f/128]) mod T, f]
//   delays[d,f]    = tanh(dm[d]*fs[f]/1000)*500
//
// Roofline: out = 512 MB write-once, x = 16 MB (L2-resident, read 32x).
// Effective HBM traffic ~528 MB -> ~23 us floor @ 23.3 TB/s. Store-bound.
// No matrix math -> WMMA inapplicable. CDNA5 paths used instead:
//   * GLOBAL_LOAD_ASYNC_TO_LDS_B128 / GLOBAL_STORE_ASYNC_FROM_LDS_B128
//     (ASYNCcnt-tracked copy engine; data never touches VGPRs)
//   * s_wait_asynccnt split-counter sync
//   * th:TH_STORE_NT on the output stream so x stays hot in the 192MB L2
// ---------------------------------------------------------------------------

#define N_DM      32
#define N_FREQ    1024
#define N_TIME    2048
#define N_POL     2
#define CHUNK     128
#define N_CHUNKS  (N_FREQ / CHUNK)   // 8
#define TGROUP    4                  // time rows per wave unit
#define ROW_BYTES 512                // 128 freq * 4B = one wave32 x 16B/lane

// Kernel 1: delays [32,1024] -> tail of d_out; per-(dm,chunk) trunc-mean
// shifts -> workspace. One block per DM, one thread per freq.
__global__ __launch_bounds__(N_FREQ)
void dedisp_delays_shifts(const float* __restrict__ dm_values,
                          const float* __restrict__ freq_scale,
                          float* __restrict__ delays_out,   // [32*1024]
                          int*   __restrict__ shifts)       // [32*8]
{
    __shared__ float sh[N_FREQ];
    const int d = blockIdx.x;
    const int f = threadIdx.x;

    const float delay = tanhf(dm_values[d] * freq_scale[f] * (1.0f / 1000.0f)) * 500.0f;
    delays_out[d * N_FREQ + f] = delay;
    sh[f] = delay;
    __syncthreads();

    if (f < N_CHUNKS) {
        float s = 0.0f;
        const int base = f * CHUNK;
        for (int i = 0; i < CHUNK; ++i) s += sh[base + i];
        // mean, trunc toward zero (delays >= 0 so trunc == floor)
        shifts[d * N_CHUNKS + f] = (int)truncf(s * (1.0f / (float)CHUNK));
    }
}

// Kernel 2: streaming gather via the CDNA5 async LDS copy engine.
// One wave32 per (dm, pol, 4-time-group, chunk): each lane moves 16B,
// wave moves one 512B chunk row per async op. global->LDS->global with
// ASYNCcnt tracking; VGPRs hold only addresses.
__global__ __launch_bounds__(256)
void dedisp_gather_async(const float* __restrict__ x,      // [2,2048,1024]
                         const int*   __restrict__ shifts, // [32,8]
                         float*       __restrict__ out)    // [32,2,2048,1024]
{
    // 8 waves/block * 4 rows * 512B = 16 KB staging, private per wave.
    __shared__ char stage[8 * TGROUP * ROW_BYTES];

    const unsigned wv   = threadIdx.x >> 5;
    const unsigned lane = threadIdx.x & 31u;
    const unsigned u    = blockIdx.x * 8u + wv;   // wave-unit id

    // u = ((d*N_POL + p) * (N_TIME/TGROUP) + tg) * N_CHUNKS + chunk
    const unsigned chunk = u & (N_CHUNKS - 1u);
    const unsigned tg    = (u >> 3) & (N_TIME / TGROUP - 1u);
    const unsigned p     = (u >> 12) & (N_POL - 1u);
    const unsigned d     =  u >> 13;

    const unsigned f4 = chunk * 32u + lane;       // 16B column within freq row

    // Wave-uniform shift -> SGPR.
    int s = shifts[d * N_CHUNKS + chunk];
    s = __builtin_amdgcn_readfirstlane(s);

    const unsigned t0      = tg * TGROUP;
    const unsigned srcPol  = p * N_TIME;
    const unsigned dstRow0 = (d * N_POL + p) * N_TIME + t0;

    // LDS byte address for this lane's 16B slot (flat-LDS addr[31:0] is the
    // DS-relative offset, per CDNA5 ISA 10.2 aperture mapping).
    const unsigned ldsBase =
        (unsigned)(uintptr_t)&stage[0] + (wv * TGROUP) * ROW_BYTES + lane * 16u;

    const unsigned long long xAddr   = (unsigned long long)(uintptr_t)x;
    const unsigned long long outAddr = (unsigned long long)(uintptr_t)out;

    // Phase 1: async global -> LDS (4 rows in flight on ASYNCcnt).
#pragma unroll
    for (int k = 0; k < TGROUP; ++k) {
        const unsigned t   = t0 + (unsigned)k;
        const unsigned st  = (t + (unsigned)s) & (N_TIME - 1u);   // roll mod 2048
        const unsigned off = (srcPol + st) * (N_FREQ * 4u) + f4 * 16u;
        const unsigned ldk = ldsBase + (unsigned)k * ROW_BYTES;
        asm volatile("global_load_async_to_lds_b128 %0, %1, %2"
                     :: "v"(ldk), "v"(off), "s"(xAddr) : "memory");
    }

    // Loads complete (LDS written) before stores read it.
    asm volatile("s_wait_asynccnt 0x0" ::: "memory");

    // Phase 2: async LDS -> global, non-temporal (don't evict x from L2).
#pragma unroll
    for (int k = 0; k < TGROUP; ++k) {
        const unsigned off = (dstRow0 + (unsigned)k) * (N_FREQ * 4u) + f4 * 16u;
        const unsigned ldk = ldsBase + (unsigned)k * ROW_BYTES;
        asm volatile("global_store_async_from_lds_b128 %0, %1, %2 th:TH_STORE_NT"
                     :: "v"(off), "v"(ldk), "s"(outAddr) : "memory");
    }
    // S_ENDPGM's implicit wait-idle drains the async stores.
}

extern "C" void kernel_launch(void* const* d_in, const int* in_sizes, int n_in,
                              void* d_out, int out_size, void* d_ws, size_t ws_size,
                              hipStream_t stream)
{
    const float* x          = (const float*)d_in[0];  // [1,2,2048,1024]
    const float* dm_values  = (const float*)d_in[1];  // [32]
    const float* freq_scale = (const float*)d_in[2];  // [1024]

    float* out        = (float*)d_out;                                          // [32,2,2048,1024]
    float* delays_out = (float*)d_out + (size_t)N_DM * N_POL * N_TIME * N_FREQ; // [32,1024]
    int*   shifts     = (int*)d_ws;                                             // 32*8 ints

    // 1) delays + shifts
    dedisp_delays_shifts<<<N_DM, N_FREQ, 0, stream>>>(dm_values, freq_scale,
                                                      delays_out, shifts);

    // 2) async streaming gather: 32 dm * 2 pol * 512 tgroups * 8 chunks units,
    //    8 wave-units per 256-thread block.
    const unsigned units  = N_DM * N_POL * (N_TIME / TGROUP) * N_CHUNKS;  // 262144
    const unsigned blocks = units / 8;                                    // 32768
    dedisp_gather_async<<<blocks, 256, 0, stream>>>(x, shifts, out);
}